// MDCA_17901423690504
// MI455X (gfx1250) — compile-verified
//
#include <hip/hip_runtime.h>
#include <hip/hip_bf16.h>

typedef __attribute__((ext_vector_type(2))) float v2f;
typedef __attribute__((ext_vector_type(8))) float v8f;

#define BATCH           32768
#define CLASSES         1000
#define CPAD            1024
#define PART_STRIDE     1032   // pad 1024 -> 1032 floats so wave-halves hit different LDS banks
#define WAVES_PER_BLOCK 8
#define MAIN_BLOCKS     512    // 4096 waves total -> 8 rows per wave

// ---------------- workspace layout (floats): [0,1024) colsum, [1024,2048) counts, [2048] ce_sum

__global__ void mdca_init(float* __restrict__ ws) {
  int i = blockIdx.x * blockDim.x + threadIdx.x;
  if (i < 2 * CPAD + 1) ws[i] = 0.0f;
}

__global__ void mdca_count(const int* __restrict__ tgt, float* __restrict__ counts) {
  int i = blockIdx.x * blockDim.x + threadIdx.x;
  if (i < BATCH) atomicAdd(&counts[tgt[i]], 1.0f);
}

__global__ __launch_bounds__(256) void mdca_main(const float* __restrict__ x,
                                                 const int* __restrict__ tgt,
                                                 float* __restrict__ colsum,
                                                 float* __restrict__ ce_sum) {
  __shared__ float part[WAVES_PER_BLOCK * PART_STRIDE];
  const int lane = threadIdx.x & 31;          // wave32
  const int warp = threadIdx.x >> 5;
  const int wave_id = blockIdx.x * WAVES_PER_BLOCK + warp;
  const int total_waves = MAIN_BLOCKS * WAVES_PER_BLOCK;

  // per-lane register-resident column partial sums: lane l owns columns l + 32k
  float acc[32];
#pragma unroll
  for (int k = 0; k < 32; ++k) acc[k] = 0.0f;
  float ce_local = 0.0f;

  for (int r = wave_id; r < BATCH; r += total_waves) {
    const float* __restrict__ xr = x + (size_t)r * CLASSES;
    if (r + total_waves < BATCH)  // speculative prefetch of next owned row (global_prefetch_b8)
      __builtin_prefetch(x + (size_t)(r + total_waves) * CLASSES + lane * 32, 0, 3);
    const int t = tgt[r];

    // pass over registers only: load row once (coalesced 128B/lane-group), row max
    float xv[32];
    float mx = -3.402823466e38f;
#pragma unroll
    for (int k = 0; k < 32; ++k) {
      const int col = lane + (k << 5);
      if (col < CLASSES) { xv[k] = xr[col]; mx = fmaxf(mx, xv[k]); }
      else xv[k] = 0.0f;
    }
#pragma unroll
    for (int off = 16; off >= 1; off >>= 1) mx = fmaxf(mx, __shfl_xor(mx, off, 32));

    // sum of exp(x - max); stash exp in xv
    float s = 0.0f;
#pragma unroll
    for (int k = 0; k < 32; ++k) {
      const int col = lane + (k << 5);
      if (col < CLASSES) { const float e = __expf(xv[k] - mx); xv[k] = e; s += e; }
    }
#pragma unroll
    for (int off = 16; off >= 1; off >>= 1) s += __shfl_xor(s, off, 32);
    const float inv = __builtin_amdgcn_rcpf(s);

    // p = softmax; accumulate column partials, t2 = sum exp(p) (for log_softmax(p)), p_target
    float tt = 0.0f, pt = 0.0f;
#pragma unroll
    for (int k = 0; k < 32; ++k) {
      const int col = lane + (k << 5);
      if (col < CLASSES) {
        const float p = xv[k] * inv;
        acc[k] += p;
        tt += __expf(p);
        if (col == t) pt = p;
      }
    }
#pragma unroll
    for (int off = 16; off >= 1; off >>= 1) {
      tt += __shfl_xor(tt, off, 32);
      pt += __shfl_xor(pt, off, 32);
    }
    const float lse = __logf(tt);
    // 0.9*(lse - p_t) + 0.1*(lse - 1/C) = lse - 0.9*p_t - 1e-4
    ce_local += lse - 0.9f * pt - 1.0e-4f;
  }

  // CE: wave reduce + one global atomic per wave
#pragma unroll
  for (int off = 16; off >= 1; off >>= 1) ce_local += __shfl_xor(ce_local, off, 32);
  if (lane == 0) atomicAdd(ce_sum, ce_local);

  // stage per-wave column partials to LDS
#pragma unroll
  for (int k = 0; k < 32; ++k) part[warp * PART_STRIDE + lane + (k << 5)] = acc[k];
  __syncthreads();

  // Cross-wave reduction via V_WMMA_F32_16X16X4_F32 with all-ones B:
  //   A[m][k] = part[wave k][column base+m]  ->  D[m][*] = sum over 4 waves of column base+m.
  // Two chained WMMAs (phases) cover all 8 waves. A layout per ISA: lanes 0-15 K=0,1; lanes 16-31 K=2,3.
  const int m  = lane & 15;
  const int hh = lane >> 4;
  v2f bones; bones.x = 1.0f; bones.y = 1.0f;
  for (int base = warp * 16; base < CPAD; base += WAVES_PER_BLOCK * 16) {
    v8f c = {};
#pragma unroll
    for (int ph = 0; ph < 2; ++ph) {
      const int w0 = ph * 4 + hh * 2;
      v2f a;
      a.x = part[w0 * PART_STRIDE + base + m];
      a.y = part[(w0 + 1) * PART_STRIDE + base + m];
      c = __builtin_amdgcn_wmma_f32_16x16x4_f32(false, a, false, bones, (short)0, c,
                                                false, false);
    }
    // D layout: lanes 0-15 VGPR i -> M=i ; lanes 16-31 VGPR i -> M=8+i (replicated over N)
    if (m == 0) {
      const int moff = hh * 8;
#pragma unroll
      for (int i = 0; i < 8; ++i) atomicAdd(&colsum[base + moff + i], c[i]);
    }
  }
}

__global__ __launch_bounds__(1024) void mdca_finalize(const float* __restrict__ colsum,
                                                      const float* __restrict__ counts,
                                                      const float* __restrict__ ce_sum,
                                                      float* __restrict__ out) {
  __shared__ float red[32];
  const int tid = threadIdx.x;
  float d = 0.0f;
  if (tid < CLASSES) d = fabsf((colsum[tid] - counts[tid]) * (1.0f / BATCH));
#pragma unroll
  for (int off = 16; off >= 1; off >>= 1) d += __shfl_xor(d, off, 32);
  if ((tid & 31) == 0) red[tid >> 5] = d;
  __syncthreads();
  if (tid < 32) {
    float v = red[tid];
#pragma unroll
    for (int off = 16; off >= 1; off >>= 1) v += __shfl_xor(v, off, 32);
    if (tid == 0) {
      const float mdca = v * (1.0f / CLASSES);
      const float ce   = ce_sum[0] * (1.0f / BATCH);
      out[0] = ce + mdca;
      out[1] = ce;
      out[2] = mdca;
    }
  }
}

extern "C" void kernel_launch(void* const* d_in, const int* in_sizes, int n_in,
                              void* d_out, int out_size, void* d_ws, size_t ws_size,
                              hipStream_t stream) {
  const float* x   = (const float*)d_in[0];
  const int*   tgt = (const int*)d_in[1];
  float* out    = (float*)d_out;
  float* colsum = (float*)d_ws;
  float* counts = colsum + CPAD;
  float* ce_sum = colsum + 2 * CPAD;

  mdca_init<<<9, 256, 0, stream>>>(colsum);
  mdca_count<<<(BATCH + 255) / 256, 256, 0, stream>>>(tgt, counts);
  mdca_main<<<MAIN_BLOCKS, 256, 0, stream>>>(x, tgt, colsum, ce_sum);
  mdca_finalize<<<1, 1024, 0, stream>>>(colsum, counts, ce_sum, out);
}